// SR_11596411699823
// MI455X (gfx1250) — compile-verified
//
#include <hip/hip_runtime.h>
#include <hip/hip_bf16.h>
#include <math.h>

// Problem constants (from reference setup_inputs)
#define NN 128      // N
#define SS 2048     // S
#define HH 256      // h
#define DD 512      // d
#define NEGV (-1000.0f)
#define ADJ (NN + SS)   // 2176

typedef __attribute__((ext_vector_type(2))) float v2f;
typedef __attribute__((ext_vector_type(8))) float v8f;

// ---------------------------------------------------------------------------
// Kernel 1: f32 WMMA GEMM.  Out[M,256] = X[M,512] @ W1[:, colOff:colOff+512].T
// One wave per 16x16 output tile; K-loop of 128 x V_WMMA_F32_16X16X4_F32.
// Grid is sized so waveCount == tileCount (EXEC all-1s for WMMA).
// ---------------------------------------------------------------------------
__global__ void gemm_wmma_f32(const float* __restrict__ X,
                              const float* __restrict__ W1,
                              int colOff,
                              float* __restrict__ Out) {
    const int wave = blockIdx.x * (blockDim.x >> 5) + (threadIdx.x >> 5);
    const int lane = threadIdx.x & 31;
    const int NT = HH / 16;              // 16 tiles along h
    const int tm = wave / NT;
    const int tn = wave % NT;
    const int m0 = tm * 16;
    const int n0 = tn * 16;

    const int r  = lane & 15;            // row within fragment
    const int kk = (lane >> 4) << 1;     // K sub-offset: lanes 0-15 -> K{0,1}, 16-31 -> K{2,3}

    const float* __restrict__ xrow = X  + (size_t)(m0 + r) * DD;        // A[m, k]
    const float* __restrict__ wrow = W1 + (size_t)(n0 + r) * (2 * DD) + colOff; // B[k,n]=W1[n,k]

    v8f c = {};
    #pragma unroll 4
    for (int k0 = 0; k0 < DD; k0 += 4) {
        v2f a, b;
        a.x = xrow[k0 + kk];
        a.y = xrow[k0 + kk + 1];
        b.x = wrow[k0 + kk];
        b.y = wrow[k0 + kk + 1];
        // 8 args: (neg_a, A, neg_b, B, c_mod, C, reuse_a, reuse_b)
        c = __builtin_amdgcn_wmma_f32_16x16x4_f32(false, a, false, b,
                                                  (short)0, c, false, false);
    }

    // C/D layout: VGPR rr -> M = m0 + rr + (lane>=16 ? 8 : 0), N = n0 + (lane&15)
    const int mhi = (lane >> 4) * 8;
    #pragma unroll
    for (int rr = 0; rr < 8; ++rr) {
        Out[(size_t)(m0 + rr + mhi) * HH + n0 + r] = c[rr];
    }
}

// ---------------------------------------------------------------------------
// Kernel 2: logits[n,s] = sum_h relu(A[n,h]+B[s,h]+b1[h]) * w2[h] + b2
// plus logsig = log_sigmoid(logits).  Each block handles one n (256 s's);
// A-row, b1, w2 staged in LDS.
// ---------------------------------------------------------------------------
__global__ void logits_kernel(const float* __restrict__ A,
                              const float* __restrict__ Bm,
                              const float* __restrict__ b1,
                              const float* __restrict__ W2,
                              const float* __restrict__ b2,
                              float* __restrict__ logits,
                              float* __restrict__ logsig) {
    __shared__ float sA[HH];
    __shared__ float sB1[HH];
    __shared__ float sW2[HH];

    const int idx = blockIdx.x * blockDim.x + threadIdx.x;   // 0 .. N*S-1
    const int n = idx >> 11;          // idx / 2048  (uniform within a block)
    const int s = idx & (SS - 1);

    const int t = threadIdx.x;
    if (t < HH) {
        sA[t]  = A[(size_t)n * HH + t];
        sB1[t] = b1[t];
        sW2[t] = W2[t];
    }
    __syncthreads();

    const float* __restrict__ brow = Bm + (size_t)s * HH;
    float acc = 0.0f;
    #pragma unroll 8
    for (int h = 0; h < HH; ++h) {
        float v = sA[h] + brow[h] + sB1[h];
        acc = fmaf(fmaxf(v, 0.0f), sW2[h], acc);
    }
    const float x = acc + b2[0];
    logits[idx] = x;
    // stable log_sigmoid
    float ls = (x >= 0.0f) ? -log1pf(expf(-x)) : (x - log1pf(expf(x)));
    logsig[idx] = ls;
}

// ---------------------------------------------------------------------------
// Kernel 3: backward DP scan over s = S-1 .. 0, single 128-thread block.
// Records take bits into take_buf[n*S + s].
// ---------------------------------------------------------------------------
__global__ void dp_kernel(const float* __restrict__ logsig,
                          unsigned char* __restrict__ take_buf) {
    __shared__ float prev[NN];
    const int n = threadIdx.x;           // 0..127
    prev[n] = NEGV;
    __syncthreads();

    for (int s = SS - 1; s >= 0; --s) {
        const float ls  = logsig[(size_t)n * SS + s];
        const float pn  = prev[n];
        const float pn1 = (n < NN - 1) ? prev[n + 1] : 0.0f;  // n==N-1 -> +0.0
        const float curr = ls + pn1;
        const bool take  = (curr >= pn);
        const float val  = take ? curr : pn;
        const bool in_range = (s >= n) && (s <= (SS - NN + n));
        const float col  = in_range ? val : NEGV;
        take_buf[(size_t)n * SS + s] = take ? 1u : 0u;
        __syncthreads();
        prev[n] = col;
        __syncthreads();
    }
}

// ---------------------------------------------------------------------------
// Kernel 4: zero the output (adj matrix + scalar slot)
// ---------------------------------------------------------------------------
__global__ void zero_kernel(float* __restrict__ p, int count) {
    const int i = blockIdx.x * blockDim.x + threadIdx.x;
    if (i < count) p[i] = 0.0f;
}

// ---------------------------------------------------------------------------
// Kernel 5: sequential walk + gather-mean + scatter ones.  Single thread;
// trivial work (2048 + 128 steps).
// ---------------------------------------------------------------------------
__global__ void finalize_kernel(const unsigned char* __restrict__ take_buf,
                                const float* __restrict__ logits,
                                float* __restrict__ out) {
    if (threadIdx.x != 0 || blockIdx.x != 0) return;
    int align[NN];
    for (int i = 0; i < NN; ++i) align[i] = 0;
    int n = 0;
    for (int s = 0; s < SS; ++s) {
        const int nc = (n < NN - 1) ? n : (NN - 1);
        const bool tc = (n < NN) && (take_buf[(size_t)nc * SS + s] != 0);
        if (tc) { align[nc] = s; ++n; }
    }
    float acc = 0.0f;
    for (int i = 0; i < NN; ++i) acc += logits[(size_t)i * SS + align[i]];
    out[(size_t)ADJ * ADJ] = acc / (float)NN;        // aggregated_logits
    for (int i = 0; i < NN; ++i) {
        out[(size_t)(SS + i) * ADJ + align[i]] = 1.0f;
    }
}

// ---------------------------------------------------------------------------
extern "C" void kernel_launch(void* const* d_in, const int* in_sizes, int n_in,
                              void* d_out, int out_size, void* d_ws, size_t ws_size,
                              hipStream_t stream) {
    const float* seg = (const float*)d_in[0];   // (128, 512)
    const float* vid = (const float*)d_in[1];   // (2048, 512)
    const float* W1  = (const float*)d_in[2];   // (256, 1024)
    const float* b1  = (const float*)d_in[3];   // (256,)
    const float* W2  = (const float*)d_in[4];   // (1, 256)
    const float* b2  = (const float*)d_in[5];   // (1,)
    float* out = (float*)d_out;                 // adj (2176*2176) + agg (1)

    float* A      = (float*)d_ws;                       // 128*256
    float* Bm     = A + NN * HH;                        // 2048*256
    float* logits = Bm + SS * HH;                       // 128*2048
    float* logsig = logits + NN * SS;                   // 128*2048
    unsigned char* take = (unsigned char*)(logsig + NN * SS); // 128*2048 bytes

    // GEMMs: one wave per 16x16 tile, 4 waves (128 thr) per block.
    // A: (128/16)*(256/16) = 128 tiles -> 32 blocks
    gemm_wmma_f32<<<32, 128, 0, stream>>>(seg, W1, 0, A);
    // B: (2048/16)*(256/16) = 2048 tiles -> 512 blocks
    gemm_wmma_f32<<<512, 128, 0, stream>>>(vid, W1, DD, Bm);

    // logits + logsig: 128*2048 threads, 256 per block (one n per block)
    logits_kernel<<<(NN * SS) / 256, 256, 0, stream>>>(A, Bm, b1, W2, b2,
                                                       logits, logsig);

    // sequential DP scan
    dp_kernel<<<1, NN, 0, stream>>>(logsig, take);

    // zero output, then walk + scatter
    const int total = ADJ * ADJ + 1;
    zero_kernel<<<(total + 255) / 256, 256, 0, stream>>>(out, total);
    finalize_kernel<<<1, 1, 0, stream>>>(take, logits, out);
}